// MSSTFF_12713103196298
// MI455X (gfx1250) — compile-verified
//
#include <hip/hip_runtime.h>
#include <hip/hip_bf16.h>

typedef __attribute__((ext_vector_type(2))) float v2f;
typedef __attribute__((ext_vector_type(4))) float v4f;
typedef __attribute__((ext_vector_type(8))) float v8f;

#define B_ 32
#define P_ 2048
#define C_ 512
#define T_ 128  // P_/16

// ---------------------------------------------------------------------------
// Pass 1: per-position channel mean via V_WMMA_F32_16X16X4_F32.
// One wave handles 16 consecutive p-positions (one 4x4 spatial tile).
// A-matrix (16x4 f32): lane l<16 holds A[M=l][K=0,1]; lane l>=16 holds
// A[M=l-16][K=2,3]  -> contiguous float2 load per lane per step.
// B-matrix filled with 1/512 so D accumulates the mean directly (f32 exact).
// ---------------------------------------------------------------------------
__global__ __launch_bounds__(256) void msstff_mean_wmma(const float* __restrict__ x,
                                                        float* __restrict__ mean) {
    const int wave = (int)((blockIdx.x * blockDim.x + threadIdx.x) >> 5);  // 0..4095
    const int lane = (int)(threadIdx.x & 31);
    const int half = lane >> 4;   // 0: K=0,1   1: K=2,3
    const int mrow = lane & 15;   // M row

    const int b    = wave >> 7;      // wave / (P_/16)
    const int tile = wave & 127;
    const int p0   = tile * 16;

    const float* rowp = x + (size_t)(b * P_ + p0 + mrow) * C_ + 2 * half;

    const float s = 1.0f / (float)C_;
    v2f bmat = {s, s};
    v8f acc  = {};

#pragma unroll 4
    for (int c0 = 0; c0 < C_; c0 += 4) {
        v2f a = *(const v2f*)(rowp + c0);
        // D = A(16x4) * B(4x16) + C ; every N column of D holds the row mean
        acc = __builtin_amdgcn_wmma_f32_16x16x4_f32(
            /*neg_a=*/false, a, /*neg_b=*/false, bmat,
            /*c_mod=*/(short)0, acc, /*reuse_a=*/false, /*reuse_b=*/false);
    }

    // C/D layout: VGPR r -> (M=r, N=lane) for lanes 0-15, (M=r+8, N=lane-16)
    // for lanes 16-31. Lane 0 holds rows 0..7, lane 16 holds rows 8..15.
    if (mrow == 0) {
        float* mp = mean + (size_t)b * P_ + p0 + 8 * half;
#pragma unroll
        for (int r = 0; r < 8; ++r) mp[r] = acc[r];
    }
}

// ---------------------------------------------------------------------------
// Pass 2: fold the three convs into one 3x3x3 stencil over (t,4,4) per batch,
// apply sigmoid. Only 65536 points -> negligible cost.
// ---------------------------------------------------------------------------
__global__ __launch_bounds__(256) void msstff_gate(const float* __restrict__ mean,
                                                   const float* __restrict__ w1,
                                                   const float* __restrict__ w2,
                                                   const float* __restrict__ w3,
                                                   float* __restrict__ sig) {
    const int idx = (int)(blockIdx.x * blockDim.x + threadIdx.x);  // 0..65535
    const int b = idx >> 11;          // / P_
    const int p = idx & (P_ - 1);
    const int z = p >> 4;
    const int y = (p >> 2) & 3;
    const int xx = p & 3;

    float g = 0.0f;
#pragma unroll
    for (int dz = -1; dz <= 1; ++dz) {
#pragma unroll
        for (int dy = -1; dy <= 1; ++dy) {
#pragma unroll
            for (int dx = -1; dx <= 1; ++dx) {
                float w = w3[((dz + 1) * 3 + (dy + 1)) * 3 + (dx + 1)];
                if (dz <= 0 && dy <= 0 && dx <= 0)
                    w += w2[((dz + 1) * 2 + (dy + 1)) * 2 + (dx + 1)];
                if (dz == 0 && dy == 0 && dx == 0)
                    w += w1[0];
                const int zz = z + dz, yy = y + dy, xq = xx + dx;
                if (zz >= 0 && zz < T_ && yy >= 0 && yy < 4 && xq >= 0 && xq < 4)
                    g += w * mean[(size_t)b * P_ + (zz << 4) + (yy << 2) + xq];
            }
        }
    }
    sig[idx] = 1.0f / (1.0f + expf(-g));
}

// ---------------------------------------------------------------------------
// Pass 3: out[b,p,c] = x[b,p,c] * sig[b,p].  Streaming, b128 loads/stores.
// x (128 MiB) should still be resident in the 192 MB L2 from pass 1.
// ---------------------------------------------------------------------------
__global__ __launch_bounds__(256) void msstff_apply(const float* __restrict__ x,
                                                    const float* __restrict__ sig,
                                                    float* __restrict__ out) {
    const size_t i4 = (size_t)blockIdx.x * blockDim.x + threadIdx.x;  // vec4 index
    const size_t p  = i4 >> 7;  // / (C_/4)
    v4f v = ((const v4f*)x)[i4];
    const float s = sig[p];
    v *= s;
    ((v4f*)out)[i4] = v;
}

extern "C" void kernel_launch(void* const* d_in, const int* in_sizes, int n_in,
                              void* d_out, int out_size, void* d_ws, size_t ws_size,
                              hipStream_t stream) {
    const float* x  = (const float*)d_in[0];
    const float* w1 = (const float*)d_in[1];
    const float* w2 = (const float*)d_in[2];
    const float* w3 = (const float*)d_in[3];
    float* out  = (float*)d_out;
    float* mean = (float*)d_ws;           // B_*P_ floats
    float* sig  = mean + (size_t)B_ * P_; // B_*P_ floats

    // Pass 1: 4096 waves (16 positions each), 8 waves per block.
    msstff_mean_wmma<<<512, 256, 0, stream>>>(x, mean);
    // Pass 2: 65536 points.
    msstff_gate<<<(B_ * P_) / 256, 256, 0, stream>>>(mean, w1, w2, w3, sig);
    // Pass 3: B_*P_*C_/4 = 8388608 vec4 elements, exact grid.
    msstff_apply<<<(B_ * P_ * (C_ / 4)) / 256, 256, 0, stream>>>(x, sig, out);
}